// SoftPattern_49220325212228
// MI455X (gfx1250) — compile-verified
//
#include <hip/hip_runtime.h>
#include <math.h>

typedef __attribute__((ext_vector_type(2))) float v2f;
typedef __attribute__((ext_vector_type(8))) float v8f;

#define WORD_DIM 300
#define PAT 6
#define TB 144        // window terms per block
#define GTILES 10     // 16-token gate tiles per block (160 gate rows incl. 16-row halo)
#define BLOCK 320     // 10 wave32s, one gate tile per wave
#define BPAD 16       // padded B row length (cols 6..15 zero)

// ---------------------------------------------------------------------------
// Tiled transpose: emb (300, VOCAB) -> embT (VOCAB, 300), so each token's
// embedding becomes a contiguous 1200 B row (contiguous v2f loads for WMMA A).
// ---------------------------------------------------------------------------
__global__ __launch_bounds__(256) void transpose_emb(
    const float* __restrict__ emb, float* __restrict__ embT, int VOCAB)
{
    __shared__ float tile[32][33];
    const int v0 = blockIdx.x * 32;
    const int d0 = blockIdx.y * 32;
    const int tx = threadIdx.x;   // 0..31
    const int ty = threadIdx.y;   // 0..7
    #pragma unroll
    for (int j = ty; j < 32; j += 8) {
        const int d = d0 + j, v = v0 + tx;
        tile[j][tx] = (d < WORD_DIM && v < VOCAB) ? emb[(long)d * VOCAB + v] : 0.f;
    }
    __syncthreads();
    #pragma unroll
    for (int j = ty; j < 32; j += 8) {
        const int v = v0 + j, d = d0 + tx;
        if (v < VOCAB && d < WORD_DIM)
            embT[(long)v * WORD_DIM + d] = tile[tx][j];
    }
}

// ---------------------------------------------------------------------------
// Main fused kernel: gate GEMM via V_WMMA_F32_16X16X4_F32 + sliding-window
// product + deterministic block reduction.
//   TR=1: embA = embT (VOCAB, 300) contiguous rows (fast path)
//   TR=0: embA = emb  (300, VOCAB) strided gather  (fallback, running pointers)
// ---------------------------------------------------------------------------
template <bool TR>
__global__ __launch_bounds__(BLOCK) void softpat_main(
    const float* __restrict__ embA,
    const float* __restrict__ diags,  // [3][300][6]
    const float* __restrict__ bias,   // [3][6]
    const int*   __restrict__ doc,    // [T]
    float* __restrict__ partials,     // [gridDim.x]
    int T, int VOCAB)
{
    __shared__ float lb[WORD_DIM * BPAD];      // zero-padded B: [300][16]
    __shared__ float lgate[GTILES * 16][PAT];  // 160 x 6 gates
    __shared__ float red[BLOCK];

    const float* diags1 = diags + WORD_DIM * PAT;  // diags[1]
    const float* bias1  = bias + PAT;              // bias[1]

    // Cooperative fill of zero-padded B (removes per-iteration cndmasks).
    for (int j = threadIdx.x; j < WORD_DIM * BPAD; j += BLOCK) {
        const int k = j >> 4, n = j & 15;
        lb[j] = (n < PAT) ? diags1[k * PAT + n] : 0.f;
    }

    const int t0   = blockIdx.x * TB;   // first term token of this block
    const int g0   = t0 - 16;           // first gate row held in LDS
    const int lane = threadIdx.x & 31;
    const int wave = threadIdx.x >> 5;
    const int m    = lane & 15;         // A-row / B-col index
    const int half = lane >> 4;

    const int tb   = g0 + wave * 16;    // this wave's gate tile base token
    const int tok  = tb + m;
    const int tokc = tok < 0 ? 0 : (tok >= T ? T - 1 : tok);
    const long v   = (long)doc[tokc];   // clamped gather column (zeroed at store)

    // Loop-invariant bases; all per-iteration addressing folds into offsets.
    const float* aptrT = embA + v * WORD_DIM + 2 * half;          // TR path
    const float* aptrG = embA + (long)(2 * half) * VOCAB + v;     // gather path
    const int    lbi   = (2 * half) * BPAD + m;                   // B base idx

    __syncthreads();  // lb ready

    v8f c = {};
    #pragma unroll
    for (int k0 = 0; k0 < WORD_DIM; k0 += 4) {
        v2f a, b;
        if constexpr (TR) {
            a = *(const v2f*)(aptrT + k0);          // 8B contiguous, offset:k0*4
        } else {
            a.x = aptrG[0];
            a.y = aptrG[VOCAB];
            aptrG += 4 * VOCAB;
        }
        b.x = lb[lbi + k0 * BPAD];                  // ds load, imm offset
        b.y = lb[lbi + (k0 + 1) * BPAD];
        c = __builtin_amdgcn_wmma_f32_16x16x4_f32(
                /*neg_a=*/false, a, /*neg_b=*/false, b,
                /*c_mod=*/(short)0, c, /*reuse_a=*/false, /*reuse_b=*/false);
    }

    // D layout: VGPR r -> tile row (r + 8*half), lane -> column n = m.
    if (m < PAT) {
        const float bb = bias1[m];
        #pragma unroll
        for (int r = 0; r < 8; ++r) {
            const int row   = r + 8 * half;
            const int token = tb + row;
            const float sg  = 1.f / (1.f + __expf(-(c[r] + bb)));
            lgate[wave * 16 + row][m] = (token >= 0 && token < T) ? sg : 0.f;
        }
    }
    __syncthreads();

    // term(t) = g[t][4]*g[t-1][3]*g[t-2][2]*g[t-3][1]*g[t-4][0]
    float term = 0.f;
    const int i = threadIdx.x;
    if (i < TB) {
        const int t = t0 + i;
        if (t < T) {
            const int li = i + 16;
            term = lgate[li - 4][0] * lgate[li - 3][1] * lgate[li - 2][2]
                 * lgate[li - 1][3] * lgate[li][4];
        }
    }
    red[threadIdx.x] = term;
    __syncthreads();
    for (int s = 256; s > 0; s >>= 1) {
        if (threadIdx.x < s && threadIdx.x + s < BLOCK)
            red[threadIdx.x] += red[threadIdx.x + s];
        __syncthreads();
    }
    if (threadIdx.x == 0) partials[blockIdx.x] = red[0];
}

__global__ __launch_bounds__(256) void softpat_finish(
    const float* __restrict__ partials, int n, float* __restrict__ out)
{
    __shared__ float red[256];
    float s = 0.f;
    for (int i = threadIdx.x; i < n; i += 256) s += partials[i];
    red[threadIdx.x] = s;
    __syncthreads();
    for (int st = 128; st > 0; st >>= 1) {
        if (threadIdx.x < st) red[threadIdx.x] += red[threadIdx.x + st];
        __syncthreads();
    }
    if (threadIdx.x == 0) out[0] = red[0];
}

extern "C" void kernel_launch(void* const* d_in, const int* in_sizes, int n_in,
                              void* d_out, int out_size, void* d_ws, size_t ws_size,
                              hipStream_t stream) {
    const float* emb   = (const float*)d_in[0];  // (300, VOCAB) f32
    const float* diags = (const float*)d_in[1];  // (3, 300, 6) f32
    const float* bias  = (const float*)d_in[2];  // (3, 6) f32
    const int*   doc   = (const int*)d_in[3];    // (T,) int
    float* out = (float*)d_out;

    const int T       = in_sizes[3];
    const int VOCAB   = in_sizes[0] / WORD_DIM;
    const int nblocks = (T + TB - 1) / TB;

    float* partials = (float*)d_ws;
    const size_t part_bytes = ((size_t)nblocks * sizeof(float) + 255) & ~(size_t)255;
    const size_t embT_bytes = (size_t)VOCAB * WORD_DIM * sizeof(float);

    if (ws_size >= part_bytes + embT_bytes) {
        // Fast path: transpose table into scratch, then contiguous WMMA A loads.
        float* embT = (float*)((char*)d_ws + part_bytes);
        dim3 tgrid((VOCAB + 31) / 32, (WORD_DIM + 31) / 32);
        transpose_emb<<<tgrid, dim3(32, 8), 0, stream>>>(emb, embT, VOCAB);
        softpat_main<true><<<nblocks, BLOCK, 0, stream>>>(embT, diags, bias, doc,
                                                          partials, T, VOCAB);
    } else {
        softpat_main<false><<<nblocks, BLOCK, 0, stream>>>(emb, diags, bias, doc,
                                                           partials, T, VOCAB);
    }
    softpat_finish<<<1, 256, 0, stream>>>(partials, nblocks, out);
}